// GCN_PYG_30812095381571
// MI455X (gfx1250) — compile-verified
//
#include <hip/hip_runtime.h>
#include <hip/hip_bf16.h>

typedef float v2f __attribute__((ext_vector_type(2)));
typedef float v8f __attribute__((ext_vector_type(8)));

#define FDIM 128
#define NGRAPHS 64

// ---------------- degree / norm ----------------

__global__ void deg_init_k(float* deg, int n) {
    int t = blockIdx.x * blockDim.x + threadIdx.x;
    if (t < n) deg[t] = 1.0f;   // self-loop weight
}

__global__ void deg_edge_k(const int* __restrict__ col, const float* __restrict__ ew,
                           float* deg, int e) {
    int t = blockIdx.x * blockDim.x + threadIdx.x;
    if (t < e) atomicAdd(&deg[col[t]], ew[t]);
}

__global__ void dinv_k(const float* __restrict__ deg, float* dinv, int n) {
    int t = blockIdx.x * blockDim.x + threadIdx.x;
    if (t < n) {
        float d = deg[t];
        dinv[t] = d > 0.0f ? rsqrtf(d) : 0.0f;
    }
}

__global__ void norm_k(const int* __restrict__ row, const int* __restrict__ col,
                       const float* __restrict__ ew, const float* __restrict__ dinv,
                       float* nrm, int e) {
    int t = blockIdx.x * blockDim.x + threadIdx.x;
    if (t < e) nrm[t] = dinv[row[t]] * ew[t] * dinv[col[t]];
}

// ---------------- dense GEMM via V_WMMA_F32_16X16X4_F32 ----------------
// C[M x 128] = A[M x 128] @ W[128 x 128]
// block = 256 threads = 8 waves; each wave computes a 16-row x 128-col stripe
// (8 tiles of 16x16), K loop in steps of 4.

__global__ __launch_bounds__(256) void gemm_wmma_k(const float* __restrict__ A,
                                                   const float* __restrict__ W,
                                                   float* __restrict__ C, int M) {
    __shared__ float lw[FDIM * FDIM];          // 64 KB, row-major W
    const int tid = threadIdx.x;
    for (int i = tid; i < FDIM * FDIM; i += 256) lw[i] = W[i];
    __syncthreads();

    const int wave = tid >> 5;
    const int lane = tid & 31;
    const int g    = lane >> 4;                // half-wave id (0/1)
    const int l16  = lane & 15;
    const int rowBase = blockIdx.x * 128 + wave * 16;

    int arow = rowBase + l16;                  // clamp (no branch: EXEC must stay all-1s)
    if (arow > M - 1) arow = M - 1;
    const float* aptr = A + (size_t)arow * FDIM;

    v8f acc[8];
#pragma unroll
    for (int t = 0; t < 8; ++t) acc[t] = (v8f){};

    for (int kk = 0; kk < FDIM; kk += 4) {
        const int ka = kk + 2 * g;             // K covered by this half-wave's 2 VGPRs
        v2f af = *(const v2f*)(aptr + ka);     // A[m][ka], A[m][ka+1]  (8B aligned)
#pragma unroll
        for (int t = 0; t < 8; ++t) {
            const int nc = t * 16 + l16;
            v2f bf;
            bf.x = lw[ka * FDIM + nc];         // W[ka][n]
            bf.y = lw[(ka + 1) * FDIM + nc];   // W[ka+1][n]
            acc[t] = __builtin_amdgcn_wmma_f32_16x16x4_f32(
                false, af, false, bf, (short)0, acc[t], false, false);
        }
    }

#pragma unroll
    for (int t = 0; t < 8; ++t) {
        const int nc = t * 16 + l16;
#pragma unroll
        for (int v = 0; v < 8; ++v) {
            const int r = rowBase + v + 8 * g; // C VGPR v -> row v + 8*g
            if (r < M) C[(size_t)r * FDIM + nc] = acc[t][v];
        }
    }
}

// ---------------- aggregation ----------------

// agg[n][:] = dinv[n]^2 * h[n][:]   (self-loop term; also serves as the zero-init)
__global__ void selfloop_init_k(const float* __restrict__ h, const float* __restrict__ dinv,
                                float* __restrict__ agg, int n) {
    int t = blockIdx.x * blockDim.x + threadIdx.x;   // n*32 threads, float4 each
    int node = t >> 5;
    if (node >= n) return;
    int j = (t & 31) * 4;
    float s = dinv[node];
    s = s * s;
    const float4 hv = *(const float4*)(h + (size_t)node * FDIM + j);
    float4 o = make_float4(s * hv.x, s * hv.y, s * hv.z, s * hv.w);
    *(float4*)(agg + (size_t)node * FDIM + j) = o;
}

// one wave (32 lanes) per edge, float4 per lane -> 128 features
__global__ __launch_bounds__(256) void edge_scatter_k(const float* __restrict__ h,
                                                      const float* __restrict__ nrm,
                                                      const int* __restrict__ row,
                                                      const int* __restrict__ col,
                                                      float* agg, int e) {
    int wid  = (blockIdx.x * 256 + threadIdx.x) >> 5;
    int lane = threadIdx.x & 31;
    if (wid >= e) return;
    int r = row[wid], c = col[wid];
    float w = nrm[wid];
    int j = lane * 4;
    const float4 hv = *(const float4*)(h + (size_t)r * FDIM + j);
    float* dst = agg + (size_t)c * FDIM + j;
    atomicAdd(dst + 0, w * hv.x);
    atomicAdd(dst + 1, w * hv.y);
    atomicAdd(dst + 2, w * hv.z);
    atomicAdd(dst + 3, w * hv.w);
}

__global__ void bias_relu_k(float* h, const float* __restrict__ b, int total) {
    int t = blockIdx.x * blockDim.x + threadIdx.x;
    if (t >= total) return;
    float v = h[t] + b[t & (FDIM - 1)];
    h[t] = v > 0.0f ? v : 0.0f;
}

// ---------------- pooling + head ----------------

__global__ void pool_init_k(float* pooled, float* counts) {
    int t = blockIdx.x * blockDim.x + threadIdx.x;
    if (t < NGRAPHS * FDIM) pooled[t] = 0.0f;
    if (t < NGRAPHS) counts[t] = 0.0f;
}

__global__ __launch_bounds__(256) void pool_scatter_k(const float* __restrict__ h,
                                                      const int* __restrict__ batch,
                                                      float* pooled, float* counts, int n) {
    int wid  = (blockIdx.x * 256 + threadIdx.x) >> 5;
    int lane = threadIdx.x & 31;
    if (wid >= n) return;
    int gph = batch[wid];
    int j = lane * 4;
    const float4 hv = *(const float4*)(h + (size_t)wid * FDIM + j);
    float* dst = pooled + gph * FDIM + j;
    atomicAdd(dst + 0, hv.x);
    atomicAdd(dst + 1, hv.y);
    atomicAdd(dst + 2, hv.z);
    atomicAdd(dst + 3, hv.w);
    if (lane == 0) atomicAdd(&counts[gph], 1.0f);
}

__global__ void head_k(const float* __restrict__ pooled, const float* __restrict__ counts,
                       const float* __restrict__ Wh, const float* __restrict__ bh,
                       float* out) {
    int gph = threadIdx.x;
    if (gph >= NGRAPHS) return;
    float inv = 1.0f / fmaxf(counts[gph], 1.0f);
    float s = 0.0f;
    for (int d = 0; d < FDIM; ++d) s += pooled[gph * FDIM + d] * Wh[d];
    out[gph] = s * inv + bh[0];
}

// ---------------- launcher ----------------

extern "C" void kernel_launch(void* const* d_in, const int* in_sizes, int n_in,
                              void* d_out, int out_size, void* d_ws, size_t ws_size,
                              hipStream_t stream) {
    const float* x   = (const float*)d_in[0];
    const float* ew  = (const float*)d_in[1];
    const float* W1  = (const float*)d_in[2];
    const float* b1  = (const float*)d_in[3];
    const float* W2  = (const float*)d_in[4];
    const float* b2  = (const float*)d_in[5];
    const float* Wh  = (const float*)d_in[6];
    const float* bh  = (const float*)d_in[7];
    const int*   ei  = (const int*)d_in[8];    // [2, E]: row then col
    const int*   bat = (const int*)d_in[9];
    float* out = (float*)d_out;

    const int N = in_sizes[0] / FDIM;          // 50000
    const int E = in_sizes[1];                 // 800000
    const int* row = ei;
    const int* col = ei + E;

    // workspace carve-up
    float* deg    = (float*)d_ws;              // N
    float* dinv   = deg + N;                   // N
    float* nrm    = dinv + N;                  // E
    float* bufA   = nrm + E;                   // N*FDIM (GEMM output / scatter source)
    float* bufB   = bufA + (size_t)N * FDIM;   // N*FDIM (aggregation target / activations)
    float* pooled = bufB + (size_t)N * FDIM;   // 64*FDIM
    float* counts = pooled + NGRAPHS * FDIM;   // 64

    const int TB = 256;
    const int gN   = (N + TB - 1) / TB;
    const int gE   = (E + TB - 1) / TB;
    const int gND  = (N * FDIM + TB - 1) / TB;
    const int gN32 = (N * 32 + TB - 1) / TB;   // wave-per-node style
    const int gEw  = (E + 7) / 8;              // wave-per-edge, 8 waves/block
    const int gNw  = (N + 7) / 8;
    const int gGemm = (N + 127) / 128;

    // normalization (shared by both layers)
    deg_init_k<<<gN, TB, 0, stream>>>(deg, N);
    deg_edge_k<<<gE, TB, 0, stream>>>(col, ew, deg, E);
    dinv_k<<<gN, TB, 0, stream>>>(deg, dinv, N);
    norm_k<<<gE, TB, 0, stream>>>(row, col, ew, dinv, nrm, E);

    // layer 1
    gemm_wmma_k<<<gGemm, TB, 0, stream>>>(x, W1, bufA, N);
    selfloop_init_k<<<gN32, TB, 0, stream>>>(bufA, dinv, bufB, N);
    edge_scatter_k<<<gEw, TB, 0, stream>>>(bufA, nrm, row, col, bufB, E);
    bias_relu_k<<<gND, TB, 0, stream>>>(bufB, b1, N * FDIM);

    // layer 2
    gemm_wmma_k<<<gGemm, TB, 0, stream>>>(bufB, W2, bufA, N);
    selfloop_init_k<<<gN32, TB, 0, stream>>>(bufA, dinv, bufB, N);
    edge_scatter_k<<<gEw, TB, 0, stream>>>(bufA, nrm, row, col, bufB, E);
    bias_relu_k<<<gND, TB, 0, stream>>>(bufB, b2, N * FDIM);

    // pooling + head
    pool_init_k<<<(NGRAPHS * FDIM + TB - 1) / TB, TB, 0, stream>>>(pooled, counts);
    pool_scatter_k<<<gNw, TB, 0, stream>>>(bufB, bat, pooled, counts, N);
    head_k<<<1, 64, 0, stream>>>(pooled, counts, Wh, bh, out);
}